// TangentProject_14224931684530
// MI455X (gfx1250) — compile-verified
//
#include <hip/hip_runtime.h>

#define PSI   0.001f
#define LDSS  65   // 64 + 1 padding to avoid LDS bank conflicts on column reads

typedef __attribute__((ext_vector_type(2))) float v2f;
typedef __attribute__((ext_vector_type(8))) float v8f;

__device__ __forceinline__ v8f wmma4(v2f a, v2f b, v8f c) {
  // D = A(16x4) * B(4x16) + C, full fp32 on the matrix pipe
  return __builtin_amdgcn_wmma_f32_16x16x4_f32(false, a, false, b, (short)0, c, false, false);
}

// ---------------------------------------------------------------------------
// Parallel cyclic Jacobi eigensolver for a symmetric 64x64 matrix in LDS.
// A is overwritten (diagonal -> eigenvalues), Q accumulates eigenvectors
// (columns). Round-robin (circle) scheduling: 63 rounds of 32 disjoint pairs.
// ---------------------------------------------------------------------------
__device__ __forceinline__ void jacobi_eigh(float (*A)[LDSS], float (*Q)[LDSS],
                                            float* pc, float* ps, int* pp, int* pq,
                                            int tid, int nthreads, int nsweeps) {
  for (int idx = tid; idx < 64 * 64; idx += nthreads) {
    int i = idx >> 6, j = idx & 63;
    Q[i][j] = (i == j) ? 1.0f : 0.0f;
  }
  __syncthreads();

  for (int sweep = 0; sweep < nsweeps; ++sweep) {
    for (int r = 0; r < 63; ++r) {
      // --- rotation angles for 32 disjoint pairs (lanes 0..31) ---
      if (tid < 32) {
        int j = tid;
        int p, q;
        if (j == 0) { p = 63; q = r % 63; }
        else        { p = (r + j) % 63; q = (r - j + 63) % 63; }
        float app = A[p][p], aqq = A[q][q], apq = A[p][q];
        float c = 1.0f, s = 0.0f;
        if (fabsf(apq) > 1e-12f) {
          float tau = (aqq - app) / (2.0f * apq);
          float t = (tau >= 0.0f ? 1.0f : -1.0f) /
                    (fabsf(tau) + sqrtf(1.0f + tau * tau));
          c = rsqrtf(1.0f + t * t);
          s = t * c;
        }
        pp[j] = p; pq[j] = q; pc[j] = c; ps[j] = s;
      }
      __syncthreads();
      // --- column rotations: A <- A*G, Q <- Q*G ---
      for (int idx = tid; idx < 32 * 64; idx += nthreads) {
        int j = idx >> 6, k = idx & 63;
        int p = pp[j], q = pq[j];
        float c = pc[j], s = ps[j];
        float ap = A[k][p], aq = A[k][q];
        A[k][p] = c * ap - s * aq;
        A[k][q] = s * ap + c * aq;
        float qp = Q[k][p], qq = Q[k][q];
        Q[k][p] = c * qp - s * qq;
        Q[k][q] = s * qp + c * qq;
      }
      __syncthreads();
      // --- row rotations: A <- G^T*A ---
      for (int idx = tid; idx < 32 * 64; idx += nthreads) {
        int j = idx >> 6, k = idx & 63;
        int p = pp[j], q = pq[j];
        float c = pc[j], s = ps[j];
        float ap = A[p][k], aq = A[q][k];
        A[p][k] = c * ap - s * aq;
        A[q][k] = s * ap + c * aq;
      }
      __syncthreads();
    }
  }
}

// ---------------------------------------------------------------------------
// Kernel 1: per weight channel c, S[c] = (W_psi)^{-1/2} via eigh + rsqrt map.
// Only C=4 matrices -> cost negligible; plain reconstruction loop.
// ---------------------------------------------------------------------------
__global__ __launch_bounds__(64)
void eigh_rsqrt_weight(const float* __restrict__ W, float* __restrict__ sri) {
  __shared__ float A[64][LDSS];
  __shared__ float Q[64][LDSS];
  __shared__ float f[64];
  __shared__ float pc[32], ps[32];
  __shared__ int   pp[32], pq[32];

  const int c   = blockIdx.x;
  const int tid = threadIdx.x;
  const float* Wc = W + (size_t)c * 4096;

  for (int idx = tid; idx < 4096; idx += 64) {
    int i = idx >> 6, j = idx & 63;
    float v = (1.0f - PSI) * Wc[idx];
    if (i == j) v += PSI;
    A[i][j] = v;
  }
  __syncthreads();

  jacobi_eigh(A, Q, pc, ps, pp, pq, tid, 64, 12);

  if (tid < 64) f[tid] = rsqrtf(fmaxf(A[tid][tid], 1e-20f));
  __syncthreads();

  // S = Q diag(f) Q^T ; one thread per output row
  const int i = tid;
  for (int j = 0; j < 64; ++j) {
    float acc = 0.0f;
    for (int k = 0; k < 64; ++k) acc += Q[i][k] * f[k] * Q[j][k];
    sri[(size_t)c * 4096 + i * 64 + j] = acc;
  }
}

// ---------------------------------------------------------------------------
// Kernel 2 (fused): per (b,c):
//   T   = S X S  (two 64^3 WMMA matmuls, symmetrized + psi reconditioning)
//   eigh(T) via Jacobi, out = Q diag(log l) Q^T (WMMA matmul, B = Q^T)
// 128 threads = 4 waves; wave w owns output row band m0 = 16w.
// ---------------------------------------------------------------------------
__global__ __launch_bounds__(128)
void tangent_project(const float* __restrict__ X, const float* __restrict__ sri,
                     float* __restrict__ out, int C) {
  __shared__ float Sm[64][LDSS];  // S, then eigenvectors Q
  __shared__ float Xm[64][LDSS];  // X, then T = S X S
  __shared__ float Mm[64][LDSS];  // M1 = S*X, then Jacobi working matrix A
  __shared__ float f[64];
  __shared__ float pc[32], ps[32];
  __shared__ int   pp[32], pq[32];

  const int bc   = blockIdx.x;
  const int b    = bc / C;
  const int c    = bc % C;
  const int tid  = threadIdx.x;
  const int lane = tid & 31;
  const int lr   = lane & 15;     // lane within half-wave
  const int hi   = lane >> 4;     // which half-wave (K/M split)
  const int m0   = (tid >> 5) * 16;

  const float* Xb = X + (size_t)b * 4096;
  const float* Sc = sri + (size_t)c * 4096;
  for (int idx = tid; idx < 4096; idx += 128) {
    Sm[idx >> 6][idx & 63] = Sc[idx];
    Xm[idx >> 6][idx & 63] = Xb[idx];
  }
  __syncthreads();

  // ---- M1 = S * X ----
  for (int n0 = 0; n0 < 64; n0 += 16) {
    v8f acc = {};
#pragma unroll
    for (int k = 0; k < 64; k += 4) {
      const int kb = k + 2 * hi;
      v2f a, bb;
      a.x  = Sm[m0 + lr][kb];     a.y  = Sm[m0 + lr][kb + 1];
      bb.x = Xm[kb][n0 + lr];     bb.y = Xm[kb + 1][n0 + lr];
      acc = wmma4(a, bb, acc);
    }
#pragma unroll
    for (int j = 0; j < 8; ++j)
      Mm[m0 + j + 8 * hi][n0 + lr] = acc[j];
  }
  __syncthreads();

  // ---- T = M1 * S  -> Xm (X no longer needed) ----
  for (int n0 = 0; n0 < 64; n0 += 16) {
    v8f acc = {};
#pragma unroll
    for (int k = 0; k < 64; k += 4) {
      const int kb = k + 2 * hi;
      v2f a, bb;
      a.x  = Mm[m0 + lr][kb];     a.y  = Mm[m0 + lr][kb + 1];
      bb.x = Sm[kb][n0 + lr];     bb.y = Sm[kb + 1][n0 + lr];
      acc = wmma4(a, bb, acc);
    }
#pragma unroll
    for (int j = 0; j < 8; ++j)
      Xm[m0 + j + 8 * hi][n0 + lr] = acc[j];
  }
  __syncthreads();

  // ---- A = (1-psi) * 0.5*(T + T^T) + psi*I  -> Mm ----
  for (int idx = tid; idx < 4096; idx += 128) {
    int i = idx >> 6, j = idx & 63;
    float v = 0.5f * (Xm[i][j] + Xm[j][i]) * (1.0f - PSI);
    if (i == j) v += PSI;
    Mm[i][j] = v;
  }
  __syncthreads();

  // ---- batched symmetric eigendecomposition (A=Mm, Q=Sm) ----
  jacobi_eigh(Mm, Sm, pc, ps, pp, pq, tid, 128, 8);

  if (tid < 64) f[tid] = logf(fmaxf(Mm[tid][tid], 1e-30f));
  __syncthreads();

  // ---- out = (Q diag f) * Q^T ; A_eff[m][k]=Q[m][k]*f[k], B_eff[k][n]=Q[n][k]
  float* outp = out + (size_t)bc * 4096;
  for (int n0 = 0; n0 < 64; n0 += 16) {
    v8f acc = {};
#pragma unroll
    for (int k = 0; k < 64; k += 4) {
      const int kb = k + 2 * hi;
      v2f a, bb;
      a.x  = Sm[m0 + lr][kb]     * f[kb];
      a.y  = Sm[m0 + lr][kb + 1] * f[kb + 1];
      bb.x = Sm[n0 + lr][kb];         // transposed read (stride-65 padded)
      bb.y = Sm[n0 + lr][kb + 1];
      acc = wmma4(a, bb, acc);
    }
#pragma unroll
    for (int j = 0; j < 8; ++j)
      outp[(m0 + j + 8 * hi) * 64 + n0 + lr] = acc[j];
  }
}

extern "C" void kernel_launch(void* const* d_in, const int* in_sizes, int n_in,
                              void* d_out, int out_size, void* d_ws, size_t ws_size,
                              hipStream_t stream) {
  (void)n_in; (void)out_size; (void)ws_size;
  const float* X = (const float*)d_in[0];   // [B,64,64]
  const float* W = (const float*)d_in[1];   // [C,64,64]
  float* out = (float*)d_out;               // [B,C,64,64]
  float* sri = (float*)d_ws;                // C*4096 floats of scratch

  const int B = in_sizes[0] / 4096;
  const int C = in_sizes[1] / 4096;

  eigh_rsqrt_weight<<<C, 64, 0, stream>>>(W, sri);
  tangent_project<<<B * C, 128, 0, stream>>>(X, sri, out, C);
}